// MutilSelfGCN_28441273434409
// MI455X (gfx1250) — compile-verified
//
#include <hip/hip_runtime.h>
#include <hip/hip_bf16.h>

// Problem constants (match reference)
#define NN 50000
#define EE 1600000
#define RR 3
#define DD 256
#define HH 8

typedef __attribute__((ext_vector_type(2))) float v2f;
typedef __attribute__((ext_vector_type(8))) float v8f;

// ---------------------------------------------------------------------------
// BatchNorm statistics: per-column sum / sum-of-squares via f32 atomics.
// ---------------------------------------------------------------------------
__global__ __launch_bounds__(256) void bn_stats_kernel(const float* __restrict__ x,
                                                       float* __restrict__ stats) {
  int d = threadIdx.x;
  float s = 0.f, s2 = 0.f;
  for (int n = blockIdx.x; n < NN; n += gridDim.x) {
    float v = x[(size_t)n * DD + d];
    s += v;
    s2 += v * v;
  }
  atomicAdd(&stats[d], s);
  atomicAdd(&stats[DD + d], s2);
}

__global__ __launch_bounds__(256) void bn_apply_kernel(const float* __restrict__ x,
                                                       const float* __restrict__ stats,
                                                       const float* __restrict__ gamma,
                                                       const float* __restrict__ beta,
                                                       float* __restrict__ xbn) {
  int d = threadIdx.x;
  size_t i = (size_t)blockIdx.x * DD + d;
  float mean = stats[d] * (1.0f / (float)NN);
  float ex2 = stats[DD + d] * (1.0f / (float)NN);
  float var = ex2 - mean * mean;
  xbn[i] = (x[i] - mean) * rsqrtf(var + 1e-5f) * gamma[d] + beta[d];
}

// ---------------------------------------------------------------------------
// GCN normalization: deg = segment_sum(w, dst); dinv = rsqrt(deg);
// norm[e] = dinv[src]*w*dinv[dst]
// ---------------------------------------------------------------------------
__global__ __launch_bounds__(256) void deg_kernel(const int* __restrict__ eidx,
                                                  const float* __restrict__ ew,
                                                  float* __restrict__ deg) {
  size_t t = (size_t)blockIdx.x * blockDim.x + threadIdx.x;
  if (t >= (size_t)RR * EE) return;
  int r = (int)(t / EE);
  int e = (int)(t % EE);
  int dst = eidx[(size_t)r * 2 * EE + EE + e];
  atomicAdd(&deg[(size_t)r * NN + dst], ew[t]);
}

__global__ __launch_bounds__(256) void dinv_kernel(float* __restrict__ dd) {
  size_t t = (size_t)blockIdx.x * blockDim.x + threadIdx.x;
  if (t >= (size_t)RR * NN) return;
  float v = dd[t];
  dd[t] = (v > 0.f) ? rsqrtf(v) : 0.f;
}

__global__ __launch_bounds__(256) void norm_kernel(const int* __restrict__ eidx,
                                                   const float* __restrict__ ew,
                                                   const float* __restrict__ dinv,
                                                   float* __restrict__ nrm) {
  size_t t = (size_t)blockIdx.x * blockDim.x + threadIdx.x;
  if (t >= (size_t)RR * EE) return;
  int r = (int)(t / EE);
  int e = (int)(t % EE);
  int src = eidx[(size_t)r * 2 * EE + e];
  int dst = eidx[(size_t)r * 2 * EE + EE + e];
  nrm[t] = dinv[(size_t)r * NN + src] * ew[t] * dinv[(size_t)r * NN + dst];
}

// Initialize SpMM accumulator with the GCN bias (broadcast over R*N rows).
__global__ __launch_bounds__(256) void bias_init_kernel(float* __restrict__ out,
                                                        const float* __restrict__ b) {
  out[(size_t)blockIdx.x * DD + threadIdx.x] = b[threadIdx.x];
}

// ---------------------------------------------------------------------------
// SpMM: out[r,dst,:] += norm * in[(r),src,:].  One wave32 per edge:
// each lane moves 8 floats (2 coalesced float4 loads + 8 f32 atomics).
// One (R,N,D) fp32 activation tensor is 154MB and fits MI455X's 192MB L2,
// so the random src-gathers are mostly L2-served.
// ---------------------------------------------------------------------------
__global__ __launch_bounds__(256) void spmm_kernel(const float* __restrict__ Hin,
                                                   const int* __restrict__ eidx,
                                                   const float* __restrict__ nrm,
                                                   float* __restrict__ Hout,
                                                   int in_rstride /*0 or N*D*/) {
  int lane = threadIdx.x & 31;
  size_t wid = (size_t)blockIdx.x * 8 + (threadIdx.x >> 5);
  if (wid >= (size_t)RR * EE) return;
  int r = (int)(wid / EE);
  int e = (int)(wid % EE);
  int src = eidx[(size_t)r * 2 * EE + e];
  int dst = eidx[(size_t)r * 2 * EE + EE + e];
  float w = nrm[(size_t)r * EE + e];
  const float* in = Hin + (size_t)r * (size_t)in_rstride + (size_t)src * DD;
  float* out = Hout + ((size_t)r * NN + dst) * DD;
  float4 a = *(const float4*)(in + lane * 4);
  float4 b = *(const float4*)(in + 128 + lane * 4);
  atomicAdd(out + lane * 4 + 0, w * a.x);
  atomicAdd(out + lane * 4 + 1, w * a.y);
  atomicAdd(out + lane * 4 + 2, w * a.z);
  atomicAdd(out + lane * 4 + 3, w * a.w);
  atomicAdd(out + 128 + lane * 4 + 0, w * b.x);
  atomicAdd(out + 128 + lane * 4 + 1, w * b.y);
  atomicAdd(out + 128 + lane * 4 + 2, w * b.z);
  atomicAdd(out + 128 + lane * 4 + 3, w * b.w);
}

// ---------------------------------------------------------------------------
// Per-node MHA over the R=3 relations. Block = node, wave = head (8 heads),
// lane = head-dim element (head_dim = 32 = wave32).
// ---------------------------------------------------------------------------
__global__ __launch_bounds__(256) void mha_kernel(const float* __restrict__ qkv,
                                                  float* __restrict__ o) {
  int n = blockIdx.x;
  int h = threadIdx.x >> 5;
  int lane = threadIdx.x & 31;
  float q[RR], k[RR], v[RR];
#pragma unroll
  for (int r = 0; r < RR; ++r) {
    size_t base = ((size_t)r * NN + n) * (3 * DD) + h * 32 + lane;
    q[r] = qkv[base];
    k[r] = qkv[base + DD];
    v[r] = qkv[base + 2 * DD];
  }
  const float scale = 0.1767766952966369f;  // 1/sqrt(32)
  float s[RR][RR];
#pragma unroll
  for (int qi = 0; qi < RR; ++qi) {
#pragma unroll
    for (int ki = 0; ki < RR; ++ki) {
      float p = q[qi] * k[ki];
#pragma unroll
      for (int off = 16; off > 0; off >>= 1) p += __shfl_xor(p, off, 32);
      s[qi][ki] = p * scale;
    }
  }
#pragma unroll
  for (int qi = 0; qi < RR; ++qi) {
    float m = fmaxf(s[qi][0], fmaxf(s[qi][1], s[qi][2]));
    float e0 = __expf(s[qi][0] - m);
    float e1 = __expf(s[qi][1] - m);
    float e2 = __expf(s[qi][2] - m);
    float inv = 1.0f / (e0 + e1 + e2);
    float ov = (e0 * v[0] + e1 * v[1] + e2 * v[2]) * inv;
    o[((size_t)qi * NN + n) * DD + h * 32 + lane] = ov;
  }
}

// ---------------------------------------------------------------------------
// WMMA GEMM: Out[m,o] = act( sum_k X[m,k]*W[o,k] + bias[o] )
// X:(M,K) row-major, W:(Dout,K) row-major.
// Block tile = 128(M) x 64(O), wave tile = 16 x 64, V_WMMA_F32_16X16X4_F32.
//
// Data movement (MI455X-tuned):
//  * W tile (64 x K) staged in LDS once per block; all 8 waves read their
//    B-fragments via ds_load (8x reuse). LDS dword index =
//    row*K + (col ^ (4*(row&15))): K is a multiple of 64, so the xor spreads
//    each wave's 64-dword read across all 64 banks with zero padding
//    (tile = 64KB for K=256, 32KB for K=128).
//  * K is a template parameter -> the k-loop fully unrolls; both the A-frag
//    (global) and the 4 B-frags (LDS) are prefetched one k-step ahead into
//    registers, so each WMMA group waits on loads issued a full step earlier
//    (s_wait_dscnt <= 4 instead of 0x0 right before every WMMA).
// Requires: K in {128,256}, M % 16 == 0, Dout % 64 == 0 -> every executing
// wave has EXEC all-ones (WMMA requirement).  act: 0=none, 1=relu, 2=leaky.
// ---------------------------------------------------------------------------
template <int K>
__global__ __launch_bounds__(256) void gemm_wmma_kernel(const float* __restrict__ X,
                                                        const float* __restrict__ W,
                                                        const float* __restrict__ bias,
                                                        float* __restrict__ Out,
                                                        int M, int Dout, int act) {
  extern __shared__ float ldsW[];  // 64 * K floats, xor-swizzled
  int tid = threadIdx.x;
  int o0 = blockIdx.y * 64;

  // Cooperative W-tile load (coalesced float4, xor-swizzled store).
  constexpr int cols4 = K >> 2;        // float4 per row
  constexpr int nvec = (64 * K) >> 2;  // total float4
#pragma unroll
  for (int idx0 = 0; idx0 < nvec; idx0 += 256) {
    int idx = idx0 + tid;
    int row = idx / cols4;
    int c4 = (idx - row * cols4) << 2;  // float col, multiple of 4
    float4 wv = *(const float4*)(W + (size_t)(o0 + row) * K + c4);
    int sc = c4 ^ ((row & 15) << 2);  // stays a multiple of 4 (16B aligned)
    *(float4*)(ldsW + row * K + sc) = wv;
  }
  __syncthreads();  // all threads reach this before any wave exits

  int lane = tid & 31;
  int wave = tid >> 5;
  int tile_m = blockIdx.x * 8 + wave;
  if (tile_m * 16 >= M) return;  // wave-uniform exit
  int m0 = tile_m * 16;
  int lr = lane & 15;
  int koff = (lane >> 4) << 1;  // 0 or 2
  int swz = lr << 2;            // xor swizzle term for B-frag reads

  const float* xp = X + (size_t)(m0 + lr) * K + koff;
  int rbK[4];
#pragma unroll
  for (int c = 0; c < 4; ++c) rbK[c] = (c * 16 + lr) * K;

  v8f acc[4];
#pragma unroll
  for (int c = 0; c < 4; ++c) acc[c] = (v8f){0.f, 0.f, 0.f, 0.f, 0.f, 0.f, 0.f, 0.f};

  // Software pipeline: A-frag (global b64) + 4 B-frags (LDS b64) prefetched
  // one k-step ahead.
  v2f a_cur, a_nxt;
  v2f b_cur[4], b_nxt[4];
  a_cur = *(const v2f*)(xp);
#pragma unroll
  for (int c = 0; c < 4; ++c)
    b_cur[c] = *(const v2f*)(ldsW + rbK[c] + (koff ^ swz));

  constexpr int NK = K / 4;
#pragma unroll
  for (int s = 0; s < NK; ++s) {
    if (s + 1 < NK) {  // compile-time resolved (full unroll)
      int kk = 4 * (s + 1);
      a_nxt = *(const v2f*)(xp + kk);
#pragma unroll
      for (int c = 0; c < 4; ++c)
        b_nxt[c] = *(const v2f*)(ldsW + rbK[c] + ((kk + koff) ^ swz));
    }
#pragma unroll
    for (int c = 0; c < 4; ++c)
      acc[c] = __builtin_amdgcn_wmma_f32_16x16x4_f32(false, a_cur, false, b_cur[c],
                                                     (short)0, acc[c], false, false);
    if (s + 1 < NK) {
      a_cur = a_nxt;
#pragma unroll
      for (int c = 0; c < 4; ++c) b_cur[c] = b_nxt[c];
    }
  }

  int rbase = m0 + ((lane >> 4) << 3);  // lanes 16-31 hold rows M=8..15
#pragma unroll
  for (int c = 0; c < 4; ++c) {
    int col = o0 + c * 16 + lr;
    float bv = bias ? bias[col] : 0.0f;
#pragma unroll
    for (int j = 0; j < 8; ++j) {
      float v = acc[c][j] + bv;
      if (act == 1) v = fmaxf(v, 0.0f);
      else if (act == 2) v = (v >= 0.0f) ? v : 0.01f * v;
      Out[(size_t)(rbase + j) * Dout + col] = v;
    }
  }
}

static void launch_gemm(hipStream_t s, const float* X, const float* W, const float* b,
                        float* O, int M, int K, int Dout, int act) {
  dim3 grid((unsigned)((M / 16 + 7) / 8), (unsigned)(Dout / 64));
  size_t shmem = (size_t)64 * K * sizeof(float);  // <= 64KB
  if (K == 256)
    gemm_wmma_kernel<256><<<grid, 256, shmem, s>>>(X, W, b, O, M, Dout, act);
  else
    gemm_wmma_kernel<128><<<grid, 256, shmem, s>>>(X, W, b, O, M, Dout, act);
}

// ---------------------------------------------------------------------------
// Orchestration
// ---------------------------------------------------------------------------
extern "C" void kernel_launch(void* const* d_in, const int* in_sizes, int n_in,
                              void* d_out, int out_size, void* d_ws, size_t ws_size,
                              hipStream_t stream) {
  const float* x = (const float*)d_in[0];
  const int* eidx = (const int*)d_in[1];
  const float* ew = (const float*)d_in[2];
  const float* gamma = (const float*)d_in[3];
  const float* beta = (const float*)d_in[4];
  const float* gcn_w = (const float*)d_in[5];
  const float* gcn_b = (const float*)d_in[6];
  const float* ain_w = (const float*)d_in[7];
  const float* ain_b = (const float*)d_in[8];
  const float* aout_w = (const float*)d_in[9];
  const float* aout_b = (const float*)d_in[10];
  const float* pw1 = (const float*)d_in[11];
  const float* pb1 = (const float*)d_in[12];
  const float* pw2 = (const float*)d_in[13];
  const float* pb2 = (const float*)d_in[14];
  float* out = (float*)d_out;

  // Workspace layout (all segments are multiples of 4 floats -> 16B aligned)
  float* ws = (float*)d_ws;
  float* xbn = ws;                                 // N*D
  float* dinv = xbn + (size_t)NN * DD;             // R*N (deg, then rsqrt in place)
  float* nrm = dinv + (size_t)RR * NN;             // R*E
  float* stats = nrm + (size_t)RR * EE;            // 2*D
  float* P = stats + 2 * DD;                       // R*N*D scratch
  float* Q = P + (size_t)RR * NN * DD;             // R*N*D activations
  float* QKV = Q + (size_t)RR * NN * DD;           // R*N*3D (also reused for proj1 out)

  // Zero only the atomically-accumulated buffers (capture-safe stream op).
  hipMemsetAsync(stats, 0, 2 * DD * sizeof(float), stream);
  hipMemsetAsync(dinv, 0, (size_t)RR * NN * sizeof(float), stream);

  // BatchNorm
  bn_stats_kernel<<<1024, 256, 0, stream>>>(x, stats);
  bn_apply_kernel<<<NN, 256, 0, stream>>>(x, stats, gamma, beta, xbn);

  // Edge normalization
  int tot_e = RR * EE;
  deg_kernel<<<(tot_e + 255) / 256, 256, 0, stream>>>(eidx, ew, dinv);
  dinv_kernel<<<(RR * NN + 255) / 256, 256, 0, stream>>>(dinv);
  norm_kernel<<<(tot_e + 255) / 256, 256, 0, stream>>>(eidx, ew, dinv, nrm);

  int spmm_blocks = (tot_e + 7) / 8;  // one wave32 per edge, 8 waves/block

  for (int i = 0; i < 3; ++i) {
    // GCN GEMM. Layer 0: xr is a broadcast of x_bn, so h = x_bn @ W^T is
    // identical across relations -> compute once (M=N), SpMM reads rstride 0.
    if (i == 0)
      launch_gemm(stream, xbn, gcn_w + (size_t)i * DD * DD, nullptr, P, NN, DD, DD, 0);
    else
      launch_gemm(stream, Q, gcn_w + (size_t)i * DD * DD, nullptr, P, RR * NN, DD, DD, 0);

    // h = spmm(h) + gcn_b : init accumulator with bias, then atomic scatter.
    bias_init_kernel<<<RR * NN, 256, 0, stream>>>(Q, gcn_b + (size_t)i * DD);
    spmm_kernel<<<spmm_blocks, 256, 0, stream>>>(P, eidx, nrm, Q, (i == 0) ? 0 : NN * DD);

    // MHA: qkv projection, per-node R x R attention, output projection (+ReLU)
    launch_gemm(stream, Q, ain_w + (size_t)i * 3 * DD * DD, ain_b + (size_t)i * 3 * DD,
                QKV, RR * NN, DD, 3 * DD, 0);
    mha_kernel<<<NN, 256, 0, stream>>>(QKV, P);
    launch_gemm(stream, P, aout_w + (size_t)i * DD * DD, aout_b + (size_t)i * DD,
                Q, RR * NN, DD, DD, /*relu*/ 1);
  }

  // Projection head (leaky relu, slope 0.01). Reuse QKV region for (R,N,128).
  launch_gemm(stream, Q, pw1, pb1, QKV, RR * NN, DD, 128, /*leaky*/ 2);
  launch_gemm(stream, QKV, pw2, pb2, out, RR * NN, 128, 64, /*leaky*/ 2);
}